// GDQN_72851235275292
// MI455X (gfx1250) — compile-verified
//
#include <hip/hip_runtime.h>
#include <hip/hip_bf16.h>
#include <math.h>

// ---------------- Types for CDNA5 WMMA ----------------
typedef __attribute__((ext_vector_type(2))) float v2f;
typedef __attribute__((ext_vector_type(8))) float v8f;

// ---------------- Problem constants ----------------
#define IN_CH     4
#define HIDDEN    64
#define N_NODES   2048
#define BGRAPH    8
#define TOTAL_N   (BGRAPH * N_NODES)        // 16384
#define N_EDGES   524288
#define MAX_NEIGH 15
#define FC1_K     (N_NODES * HIDDEN)        // 131072
#define FC1_N     N_NODES                   // 2048

// SELU constants
#define SELU_SCALE 1.0507009873554805f
#define SELU_ALPHA 1.6732632423543772f

__device__ __forceinline__ float selu(float x) {
    return SELU_SCALE * (x > 0.0f ? x : SELU_ALPHA * (expf(x) - 1.0f));
}

// ---------------- Degree / normalization ----------------
__global__ void k_deg_init(float* deg) {
    int i = blockIdx.x * blockDim.x + threadIdx.x;
    if (i < TOTAL_N) deg[i] = 1.0f;   // self-loop
}

__global__ void k_deg_edges(const long long* dst, float* deg) {
    int e = blockIdx.x * blockDim.x + threadIdx.x;
    if (e < N_EDGES) atomicAdd(&deg[(int)dst[e]], 1.0f);
}

__global__ void k_dinv(const float* deg, float* dinv) {
    int i = blockIdx.x * blockDim.x + threadIdx.x;
    if (i < TOTAL_N) dinv[i] = rsqrtf(deg[i]);   // deg >= 1 always (self-loop)
}

// ---------------- Layer-1 feature transform: hpre = x @ W1 ----------------
__global__ void k_xw1(const float* __restrict__ x, const float* __restrict__ W1,
                      float* __restrict__ hpre) {
    int idx = blockIdx.x * blockDim.x + threadIdx.x;
    if (idx >= TOTAL_N * HIDDEN) return;
    int i = idx >> 6, j = idx & 63;
    float s = 0.0f;
#pragma unroll
    for (int k = 0; k < IN_CH; ++k)
        s = fmaf(x[i * IN_CH + k], W1[k * HIDDEN + j], s);
    hpre[idx] = s;
}

// ---------------- Aggregation: init with self-loop term ----------------
__global__ void k_agg_init(const float* __restrict__ hpre, const float* __restrict__ dinv,
                           float* __restrict__ agg) {
    int idx = blockIdx.x * blockDim.x + threadIdx.x;
    if (idx >= TOTAL_N * HIDDEN) return;
    int i = idx >> 6;
    float di = dinv[i];
    agg[idx] = hpre[idx] * di * di;
}

// ---------------- Aggregation: edge scatter (one wave per edge, 2 ch/lane) --
__global__ void k_agg_edges(const long long* __restrict__ src, const long long* __restrict__ dst,
                            const float* __restrict__ hpre, const float* __restrict__ dinv,
                            float* __restrict__ agg) {
    int t = blockIdx.x * blockDim.x + threadIdx.x;
    int e = t >> 5;
    if (e >= N_EDGES) return;
    int lane = t & 31;
    int s = (int)src[e];
    int d = (int)dst[e];
    float norm = dinv[s] * dinv[d];
    int c = lane * 2;
    atomicAdd(&agg[d * HIDDEN + c],     hpre[s * HIDDEN + c]     * norm);
    atomicAdd(&agg[d * HIDDEN + c + 1], hpre[s * HIDDEN + c + 1] * norm);
}

// ---------------- Bias + SELU ----------------
__global__ void k_bias_selu(const float* __restrict__ agg, const float* __restrict__ bias,
                            float* __restrict__ h) {
    int idx = blockIdx.x * blockDim.x + threadIdx.x;
    if (idx >= TOTAL_N * HIDDEN) return;
    h[idx] = selu(agg[idx] + bias[idx & 63]);
}

// ---------------- Layer-2 GEMM via native f32 WMMA: hpre2 = h1 @ W2 --------
// M = 16384, K = 64, N = 64.  One wave per 16x16 tile: 16x V_WMMA_F32_16X16X4_F32.
// f32 A fragment layout (ISA 7.12.2): lane m = lane&15, hi = lane>>4,
//   a = { A[m][k0+2*hi], A[m][k0+2*hi+1] }  -> one aligned b64 load.
__global__ __launch_bounds__(256) void k_gemm_w2(const float* __restrict__ h1,
                                                 const float* __restrict__ W2,
                                                 float* __restrict__ hpre2) {
    int wave  = blockIdx.x * (blockDim.x >> 5) + (threadIdx.x >> 5);
    int mtile = wave >> 2;       // 0..1023
    int ntile = wave & 3;        // 0..3
    int lane  = threadIdx.x & 31;
    int m16   = lane & 15;
    int hi    = lane >> 4;
    int mrow  = mtile * 16 + m16;
    int ncol  = ntile * 16 + m16;

    v8f acc = {};
#pragma unroll
    for (int k0 = 0; k0 < HIDDEN; k0 += 4) {
        int ka = k0 + 2 * hi;
        v2f a = *(const v2f*)&h1[mrow * HIDDEN + ka];          // 8B aligned
        v2f b;
        b[0] = W2[ka * HIDDEN + ncol];
        b[1] = W2[(ka + 1) * HIDDEN + ncol];
        acc = __builtin_amdgcn_wmma_f32_16x16x4_f32(false, a, false, b,
                                                    (short)0, acc, false, false);
    }
#pragma unroll
    for (int r = 0; r < 8; ++r) {
        int row = mtile * 16 + r + hi * 8;
        hpre2[row * HIDDEN + ncol] = acc[r];
    }
}

// ---------------- FC1 via f32 WMMA: acc[8,2048] += h[8,131072] @ Wf1 -------
// Bandwidth-bound stream of Wf1 (1.07 GB f32, read exactly once).
// K split into 64 chunks of 2048, N into 128 tiles of 16 -> 8192 waves,
// f32 atomic reduction into [8,2048].  A rows 8..15 duplicate rows 0..7
// (their D rows are discarded), so EXEC stays all-1s at every WMMA.
#define FC1_ITERS 64            // 64 * 32 = 2048 K per chunk
__global__ __launch_bounds__(256) void k_fc1_wmma(const float* __restrict__ h,
                                                  const float* __restrict__ Wf1,
                                                  float* __restrict__ fc1acc) {
    int wave   = blockIdx.x * (blockDim.x >> 5) + (threadIdx.x >> 5);
    int ntile  = wave & 127;
    int kchunk = wave >> 7;
    int lane   = threadIdx.x & 31;
    int hi     = lane >> 4;
    int mA     = lane & 7;                 // duplicate rows for M=8..15
    int ncol   = ntile * 16 + (lane & 15);

    v8f acc = {};
    for (int t = 0; t < FC1_ITERS; ++t) {
        int kbase = kchunk * (FC1_ITERS * 32) + t * 32;

        // stream-ahead prefetch of the Wf1 tile two iterations out
        int pk = kbase + 64; if (pk > FC1_K - 1) pk = FC1_K - 1;
        __builtin_prefetch(&Wf1[(size_t)pk * FC1_N + ncol], 0, 0);

#pragma unroll
        for (int kk = 0; kk < 32; kk += 4) {
            int ka = kbase + kk + 2 * hi;
            v2f a = *(const v2f*)&h[(size_t)mA * FC1_K + ka];  // 8B aligned
            v2f b;
            size_t bi = (size_t)ka * FC1_N + ncol;
            b[0] = Wf1[bi];
            b[1] = Wf1[bi + FC1_N];
            acc = __builtin_amdgcn_wmma_f32_16x16x4_f32(false, a, false, b,
                                                        (short)0, acc, false, false);
        }
    }
    if (hi == 0) {               // lanes 0..15 hold D rows 0..7 -> the 8 live rows
#pragma unroll
        for (int r = 0; r < 8; ++r)
            atomicAdd(&fc1acc[r * FC1_N + ncol], acc[r]);
    }
}

__global__ void k_zero(float* p, int n) {
    int i = blockIdx.x * blockDim.x + threadIdx.x;
    if (i < n) p[i] = 0.0f;
}

__global__ void k_fc1_post(const float* __restrict__ acc, const float* __restrict__ bf1,
                           float* __restrict__ fc1) {
    int idx = blockIdx.x * blockDim.x + threadIdx.x;
    if (idx >= BGRAPH * FC1_N) return;
    fc1[idx] = selu(acc[idx] + bf1[idx & (FC1_N - 1)]);
}

// ---------------- FC2: out[8,15] = fc1 @ Wf2 + bf2 ----------------
__global__ void k_fc2(const float* __restrict__ fc1, const float* __restrict__ Wf2,
                      const float* __restrict__ bf2, float* __restrict__ out) {
    __shared__ float red[256];
    int bg = blockIdx.x / MAX_NEIGH;
    int mc = blockIdx.x % MAX_NEIGH;
    float s = 0.0f;
    for (int n = threadIdx.x; n < FC1_N; n += 256)
        s = fmaf(fc1[bg * FC1_N + n], Wf2[n * MAX_NEIGH + mc], s);
    red[threadIdx.x] = s;
    __syncthreads();
    for (int off = 128; off > 0; off >>= 1) {
        if ((int)threadIdx.x < off) red[threadIdx.x] += red[threadIdx.x + off];
        __syncthreads();
    }
    if (threadIdx.x == 0) out[bg * MAX_NEIGH + mc] = red[0] + bf2[mc];
}

// =====================================================================
extern "C" void kernel_launch(void* const* d_in, const int* in_sizes, int n_in,
                              void* d_out, int out_size, void* d_ws, size_t ws_size,
                              hipStream_t stream) {
    const float*     x    = (const float*)d_in[0];
    const long long* eidx = (const long long*)d_in[1];   // int64 [2, E]
    const float*     W1   = (const float*)d_in[2];
    const float*     b1   = (const float*)d_in[3];
    const float*     W2   = (const float*)d_in[4];
    const float*     b2   = (const float*)d_in[5];
    const float*     Wf1  = (const float*)d_in[6];
    const float*     bf1  = (const float*)d_in[7];
    const float*     Wf2  = (const float*)d_in[8];
    const float*     bf2  = (const float*)d_in[9];
    float*           out  = (float*)d_out;

    const long long* src = eidx;
    const long long* dst = eidx + N_EDGES;

    // Workspace carve-up (~12.8 MB)
    float* deg    = (float*)d_ws;
    float* dinv   = deg  + TOTAL_N;
    float* hpre   = dinv + TOTAL_N;                    // [16384*64]
    float* agg    = hpre + TOTAL_N * HIDDEN;           // [16384*64]
    float* hf32   = agg  + TOTAL_N * HIDDEN;           // [16384*64] (h1, then h2)
    float* fc1acc = hf32 + TOTAL_N * HIDDEN;           // [8*2048]
    float* fc1    = fc1acc + BGRAPH * FC1_N;           // [8*2048]

    const int NB = 256;
    const int nodes_b = (TOTAL_N + NB - 1) / NB;                 // 64
    const int feat_b  = (TOTAL_N * HIDDEN + NB - 1) / NB;        // 4096
    const int edge_b  = (N_EDGES + NB - 1) / NB;                 // 2048
    const int escb    = (N_EDGES * 32) / NB;                     // 65536

    // ---- degree / normalization (shared by both GCN layers) ----
    k_deg_init <<<nodes_b, NB, 0, stream>>>(deg);
    k_deg_edges<<<edge_b,  NB, 0, stream>>>(dst, deg);
    k_dinv     <<<nodes_b, NB, 0, stream>>>(deg, dinv);

    // ---- GCN layer 1 ----
    k_xw1      <<<feat_b, NB, 0, stream>>>(x, W1, hpre);
    k_agg_init <<<feat_b, NB, 0, stream>>>(hpre, dinv, agg);
    k_agg_edges<<<escb,   NB, 0, stream>>>(src, dst, hpre, dinv, agg);
    k_bias_selu<<<feat_b, NB, 0, stream>>>(agg, b1, hf32);       // hf32 = h1

    // ---- GCN layer 2: native f32 WMMA GEMM then aggregation ----
    k_gemm_w2  <<<512, NB, 0, stream>>>(hf32, W2, hpre);         // 4096 waves
    k_agg_init <<<feat_b, NB, 0, stream>>>(hpre, dinv, agg);
    k_agg_edges<<<escb,   NB, 0, stream>>>(src, dst, hpre, dinv, agg);
    k_bias_selu<<<feat_b, NB, 0, stream>>>(agg, b2, hf32);       // hf32 = h[8,131072]

    // ---- FC1: bandwidth-bound f32 WMMA stream over 1.07 GB Wf1 ----
    k_zero    <<<(BGRAPH * FC1_N + NB - 1) / NB, NB, 0, stream>>>(fc1acc, BGRAPH * FC1_N);
    k_fc1_wmma<<<1024, NB, 0, stream>>>(hf32, Wf1, fc1acc);      // 8192 waves
    k_fc1_post<<<(BGRAPH * FC1_N + NB - 1) / NB, NB, 0, stream>>>(fc1acc, bf1, fc1);

    // ---- FC2 ----
    k_fc2<<<BGRAPH * MAX_NEIGH, NB, 0, stream>>>(fc1, Wf2, bf2, out);
}